// HyperConnections_59253368815636
// MI455X (gfx1250) — compile-verified
//
#include <hip/hip_runtime.h>
#include <math.h>

typedef float v2f __attribute__((ext_vector_type(2)));
typedef float v4f __attribute__((ext_vector_type(4)));
typedef float v8f __attribute__((ext_vector_type(8)));

#define SINKHORN_ITERS 10
#define SINKHORN_TAU 0.05f
#define UNROLL 8

// ---------------------------------------------------------------------------
// Prep: fold Sinkhorn(H_res) + softmax(H_pre) + softmax(H_post) into a single
// effective 4x4 mixing matrix W[s_out][s_in], written to d_ws (16 floats).
// ---------------------------------------------------------------------------
__global__ void hc_prep_kernel(const float* __restrict__ hres_logits,
                               const float* __restrict__ hpre_logits,
                               const float* __restrict__ hpost_logits,
                               float* __restrict__ w) {
    if (threadIdx.x != 0 || blockIdx.x != 0) return;

    float Z[4][4], u[4], v[4];
    for (int i = 0; i < 4; ++i)
        for (int j = 0; j < 4; ++j)
            Z[i][j] = hres_logits[i * 4 + j] * (1.0f / SINKHORN_TAU);
    const float logm = -logf(4.0f);
    for (int i = 0; i < 4; ++i) { u[i] = 0.0f; v[i] = 0.0f; }

    for (int it = 0; it < SINKHORN_ITERS; ++it) {
        for (int i = 0; i < 4; ++i) {
            float m = -__builtin_inff();
            for (int j = 0; j < 4; ++j) m = fmaxf(m, Z[i][j] + v[j]);
            float s = 0.0f;
            for (int j = 0; j < 4; ++j) s += expf(Z[i][j] + v[j] - m);
            u[i] = logm - (m + logf(s));
        }
        for (int j = 0; j < 4; ++j) {
            float m = -__builtin_inff();
            for (int i = 0; i < 4; ++i) m = fmaxf(m, Z[i][j] + u[i]);
            float s = 0.0f;
            for (int i = 0; i < 4; ++i) s += expf(Z[i][j] + u[i] - m);
            v[j] = logm - (m + logf(s));
        }
    }

    float Hres[4][4];
    for (int i = 0; i < 4; ++i)
        for (int j = 0; j < 4; ++j)
            Hres[i][j] = expf(Z[i][j] + u[i] + v[j]) * 4.0f;

    float pre[4], post[4];
    {
        float m = -__builtin_inff();
        for (int i = 0; i < 4; ++i) m = fmaxf(m, hpre_logits[i]);
        float s = 0.0f;
        for (int i = 0; i < 4; ++i) { pre[i] = expf(hpre_logits[i] - m); s += pre[i]; }
        for (int i = 0; i < 4; ++i) pre[i] /= s;
    }
    {
        float m = -__builtin_inff();
        for (int i = 0; i < 4; ++i) m = fmaxf(m, hpost_logits[i]);
        float s = 0.0f;
        for (int i = 0; i < 4; ++i) { post[i] = expf(hpost_logits[i] - m); s += post[i]; }
        for (int i = 0; i < 4; ++i) post[i] /= s;
    }

    for (int so = 0; so < 4; ++so)
        for (int si = 0; si < 4; ++si)
            w[so * 4 + si] = Hres[si][so] + post[so] * pre[si];
}

// ---------------------------------------------------------------------------
// Main streaming mix. One V_WMMA_F32_16X16X4_F32 per 16-position tile.
//   A (16x4): lane -> pos=lane%16, K={2*(lane/16), +1}: two b32 NT loads,
//             both saddr-form against ONE uniform base with two 32-bit
//             voffset registers (stream shift L*4 folded into voff_in1).
//   B (4x16): W^T broadcast into columns 0..3 (built once, lives in 2 VGPRs).
//   D: lanes 0..3 own pos 0..7, lanes 16..19 own pos 8..15 of stream lane&3
//      -> two b128 NT stores per active lane, imm-offset addressed.
// UNROLL=8 tiles per outer iteration: 16 loads clause'd ahead of the 8
// wmma+store groups (~2KB in flight per wave); voffsets bump once per group
// so the unrolled body uses only instruction immediate offsets.
// ---------------------------------------------------------------------------
__global__ void __launch_bounds__(256) hc_mix_wmma_kernel(
        const float* __restrict__ in,
        const float* __restrict__ w,
        float* __restrict__ out,
        long long L,            // per-(b,s) plane length = T*D
        int tilesPerB,          // L / 16
        int tilesPerWave,
        int gridPerB) {
    const int lane      = threadIdx.x & 31;
    const int waveInBlk = threadIdx.x >> 5;
    const int b         = (blockIdx.x >= (unsigned)gridPerB) ? 1 : 0;
    const int blkInB    = (int)blockIdx.x - b * gridPerB;
    const int wIdx      = blkInB * ((int)blockDim.x >> 5) + waveInBlk;

    long long t0 = (long long)wIdx * tilesPerWave;
    long long t1 = t0 + tilesPerWave;
    if (t1 > tilesPerB) t1 = tilesPerB;
    if (t0 >= t1) return;                       // wave-uniform

    const int m    = lane & 15;
    const int half = lane >> 4;
    const int nc   = lane & 3;
    const bool storer = (lane & 12) == 0;       // lanes 0..3 and 16..19

    v2f bmat;
    bmat.x = w[nc * 4 + 2 * half + 0];
    bmat.y = w[nc * 4 + 2 * half + 1];

    const char* pin  = (const char*)(in + (long long)b * 4 * L);
    char*       pout = (char*)(out + (long long)b * 4 * L);

    // 32-bit byte offsets (max ~128MB, fits easily).
    const unsigned dL = (unsigned)(L * 4);      // uniform stream shift
    unsigned voff_in0 = (unsigned)(((long long)(2 * half) * L + t0 * 16 + m) * 4);
    unsigned voff_in1 = voff_in0 + dL;
    unsigned voff_out = (unsigned)(((long long)nc * L + t0 * 16 + (half << 3)) * 4);

    long long n      = t1 - t0;
    long long nOuter = n / UNROLL;
    int       nTail  = (int)(n - nOuter * UNROLL);

    for (long long it = 0; it < nOuter; ++it) {
        float ax[UNROLL], ay[UNROLL];
#pragma unroll
        for (int u = 0; u < UNROLL; ++u) {
            ax[u] = __builtin_nontemporal_load((const float*)(pin + voff_in0 + u * 64));
            ay[u] = __builtin_nontemporal_load((const float*)(pin + voff_in1 + u * 64));
        }
#pragma unroll
        for (int u = 0; u < UNROLL; ++u) {
            v2f amat = {ax[u], ay[u]};
            v8f c = {};
            c = __builtin_amdgcn_wmma_f32_16x16x4_f32(
                    false, amat, false, bmat, (short)0, c, false, false);
            if (storer) {
                v4f lo = {c[0], c[1], c[2], c[3]};
                v4f hi = {c[4], c[5], c[6], c[7]};
                __builtin_nontemporal_store(lo, (v4f*)(pout + voff_out + u * 64));
                __builtin_nontemporal_store(hi, (v4f*)(pout + voff_out + u * 64 + 16));
            }
        }
        voff_in0 += 64 * UNROLL;
        voff_in1 += 64 * UNROLL;
        voff_out += 64 * UNROLL;
    }

    for (int it = 0; it < nTail; ++it) {
        v2f amat;
        amat.x = __builtin_nontemporal_load((const float*)(pin + voff_in0));
        amat.y = __builtin_nontemporal_load((const float*)(pin + voff_in1));
        v8f c = {};
        c = __builtin_amdgcn_wmma_f32_16x16x4_f32(
                false, amat, false, bmat, (short)0, c, false, false);
        if (storer) {
            v4f lo = {c[0], c[1], c[2], c[3]};
            v4f hi = {c[4], c[5], c[6], c[7]};
            __builtin_nontemporal_store(lo, (v4f*)(pout + voff_out));
            __builtin_nontemporal_store(hi, (v4f*)(pout + voff_out + 16));
        }
        voff_in0 += 64;
        voff_in1 += 64;
        voff_out += 64;
    }
}

extern "C" void kernel_launch(void* const* d_in, const int* in_sizes, int n_in,
                              void* d_out, int out_size, void* d_ws, size_t ws_size,
                              hipStream_t stream) {
    const float* resid = (const float*)d_in[0];  // (B*S, T, D) fp32
    const float* hres  = (const float*)d_in[1];  // (4,4)
    const float* hpre  = (const float*)d_in[2];  // (4,)
    const float* hpost = (const float*)d_in[3];  // (4,)
    float* out = (float*)d_out;
    float* w   = (float*)d_ws;                   // 16 floats scratch

    hc_prep_kernel<<<1, 32, 0, stream>>>(hres, hpre, hpost, w);

    const long long B = 2, S = 4;
    const long long total = (long long)in_sizes[0];
    const long long L = total / (B * S);           // T*D = 8M for reference
    const int tilesPerB = (int)(L / 16);           // 524288

    const int block = 256;                         // 8 waves (wave32)
    const int wavesPerBlock = block / 32;

    // 64 tiles per wave (8 outer iterations of UNROLL=8).
    int wavesPerB = tilesPerB / 64;
    if (wavesPerB < 1) wavesPerB = 1;
    int gridPerB = (wavesPerB + wavesPerBlock - 1) / wavesPerBlock;   // 1024
    wavesPerB = gridPerB * wavesPerBlock;                              // 8192
    int tilesPerWave = (tilesPerB + wavesPerB - 1) / wavesPerB;        // 64

    hc_mix_wmma_kernel<<<2 * gridPerB, block, 0, stream>>>(
        resid, w, out, L, tilesPerB, tilesPerWave, gridPerB);
}